// LoRARowParallelLinear_22101901705624
// MI455X (gfx1250) — compile-verified
//
#include <hip/hip_runtime.h>
#include <hip/hip_bf16.h>

// ---------------------------------------------------------------------------
// LoRA row-parallel linear for MI455X (gfx1250, wave32).
//   y = x @ W^T + bias + scatter-add( scale[slot] * (x @ A[slot]^T) @ B[slot]^T )
// fp32 end-to-end using V_WMMA_F32_16X16X4_F32; tiles fed through the CDNA5
// async global->LDS engine (ASYNCcnt) with distance-2 global_prefetch.
// ---------------------------------------------------------------------------

typedef __attribute__((ext_vector_type(2))) float v2f;
typedef __attribute__((ext_vector_type(8))) float v8f;
typedef __attribute__((ext_vector_type(4))) int   v4i;

typedef __attribute__((address_space(1))) v4i glob_v4i;   // global int4
typedef __attribute__((address_space(3))) v4i lds_v4i;    // LDS int4

#define TDIM   8192
#define DIN    4096
#define DOUT   4096
#define NSLOT  8
#define RANK   16

#define BM 128          // block tile rows (tokens)
#define BN 128          // block tile cols (outputs)
#define KT 16           // K slice per LDS stage
#define NK (DIN / KT)   // 256 stages
#define LSTR 20         // LDS row stride in floats: 16B-aligned, conflict-free

#if defined(__has_builtin)
#  if __has_builtin(__builtin_amdgcn_global_load_async_to_lds_b128)
#    define HAVE_ASYNC_LDS 1
#  endif
#endif
#ifndef HAVE_ASYNC_LDS
#  define HAVE_ASYNC_LDS 0
#endif

__device__ __forceinline__ void wait_async0() {
#if defined(__has_builtin) && __has_builtin(__builtin_amdgcn_s_wait_asynccnt)
    __builtin_amdgcn_s_wait_asynccnt(0);
#else
    asm volatile("s_wait_asynccnt 0x0" ::: "memory");
#endif
}

#if HAVE_ASYNC_LDS
// 16B async copy global -> LDS (per-lane addresses, tracked by ASYNCcnt).
// Builtin signature: (int4 AS1*, int4 AS3*, imm offset, imm cpol).
// Low 32 bits of a generic LDS pointer are the LDS byte offset on CDNA5.
__device__ __forceinline__ void cp_async_16(const float* g, float* l) {
    __builtin_amdgcn_global_load_async_to_lds_b128(
        (glob_v4i*)(uintptr_t)g,
        (lds_v4i*)(uint32_t)(uintptr_t)l,
        0, 0);
}
#endif

// ---------------------------------------------------------------------------
// Kernel A: u[t, r] = scale[slot(t)] * sum_d x[t,d] * lora_A[slot(t), r, d]
// One wave32 per token; x row is read exactly once; lora_A (2 MB) stays in L2.
// ---------------------------------------------------------------------------
__global__ __launch_bounds__(256) void lora_u_kernel(
    const float* __restrict__ x, const int* __restrict__ t2s,
    const float* __restrict__ loraA, const float* __restrict__ scaling,
    float* __restrict__ u)
{
    const int wave = threadIdx.x >> 5;
    const int lane = threadIdx.x & 31;
    const int t    = blockIdx.x * 8 + wave;

    int slot = t2s[t];
    if (slot >= NSLOT) slot = -1;   // safety

    float acc[RANK];
#pragma unroll
    for (int r = 0; r < RANK; ++r) acc[r] = 0.f;

    if (slot >= 0) {
        const float* __restrict__ xr = x + (size_t)t * DIN;
        const float* __restrict__ Ar = loraA + (size_t)slot * RANK * DIN;
        for (int i = 0; i < DIN / 128; ++i) {          // 32 iters, float4 per lane
            const int d = (i * 32 + lane) * 4;
            const float4 xv = *(const float4*)(xr + d);
#pragma unroll
            for (int r = 0; r < RANK; ++r) {
                const float4 av = *(const float4*)(Ar + r * DIN + d);
                acc[r] += xv.x * av.x + xv.y * av.y + xv.z * av.z + xv.w * av.w;
            }
        }
    }

    // wave32 butterfly reduction for each rank
#pragma unroll
    for (int r = 0; r < RANK; ++r) {
        float v = acc[r];
#pragma unroll
        for (int off = 16; off >= 1; off >>= 1) v += __shfl_xor(v, off, 32);
        acc[r] = v;
    }

    if (lane == 0) {
        const float s = (slot >= 0) ? scaling[slot] : 0.f;
        float4* up = (float4*)(u + (size_t)t * RANK);
#pragma unroll
        for (int q = 0; q < 4; ++q)
            up[q] = make_float4(acc[4*q]*s, acc[4*q+1]*s, acc[4*q+2]*s, acc[4*q+3]*s);
    }
}

// ---------------------------------------------------------------------------
// Kernel B: y = x @ W^T via v_wmma_f32_16x16x4_f32, fused bias + LoRA epilogue.
// 256 threads = 8 waves; block tile 128x128; wave tile 64x32 (4x2 fragments).
// Double-buffered LDS fed by async global->LDS copies; prefetch distance 2.
// ---------------------------------------------------------------------------
__global__ __launch_bounds__(256) void lora_gemm_f32_wmma(
    const float* __restrict__ x, const int* __restrict__ t2s,
    const float* __restrict__ w, const float* __restrict__ bias,
    const float* __restrict__ loraB, const float* __restrict__ u,
    float* __restrict__ out)
{
    __shared__ float xs[2][BM][LSTR];   // 20480 B
    __shared__ float ws[2][BN][LSTR];   // 20480 B

    const int tid   = threadIdx.x;
    const int oTile = blockIdx.x * BN;
    const int tTile = blockIdx.y * BM;

    const int wave = tid >> 5;
    const int lane = tid & 31;
    const int wm   = wave & 1;          // 2 waves along M
    const int wn   = wave >> 1;         // 4 waves along N
    const int half = lane >> 4;         // ISA 16x16x4 f32 fragment layout
    const int idx  = lane & 15;
    const int mBase = wm * 64;
    const int nBase = wn * 32;

    // global->LDS staging: 512 x 16B per (128x16) tile, 2 rows per thread
    const int r0 = tid >> 2;            // 0..63
    const int r1 = r0 + 64;             // 64..127
    const int c4 = (tid & 3) * 4;       // 0,4,8,12

    const float* __restrict__ xg0 = x + (size_t)(tTile + r0) * DIN + c4;
    const float* __restrict__ xg1 = x + (size_t)(tTile + r1) * DIN + c4;
    const float* __restrict__ wg0 = w + (size_t)(oTile + r0) * DIN + c4;
    const float* __restrict__ wg1 = w + (size_t)(oTile + r1) * DIN + c4;

#if HAVE_ASYNC_LDS
#define LOAD_STAGE(buf, kp)                              \
    do {                                                 \
        cp_async_16(xg0 + (kp), &xs[buf][r0][c4]);       \
        cp_async_16(xg1 + (kp), &xs[buf][r1][c4]);       \
        cp_async_16(wg0 + (kp), &ws[buf][r0][c4]);       \
        cp_async_16(wg1 + (kp), &ws[buf][r1][c4]);       \
    } while (0)
#define WAIT_STAGE() wait_async0()
#else
#define LOAD_STAGE(buf, kp)                                              \
    do {                                                                 \
        *(float4*)(&xs[buf][r0][c4]) = *(const float4*)(xg0 + (kp));     \
        *(float4*)(&xs[buf][r1][c4]) = *(const float4*)(xg1 + (kp));     \
        *(float4*)(&ws[buf][r0][c4]) = *(const float4*)(wg0 + (kp));     \
        *(float4*)(&ws[buf][r1][c4]) = *(const float4*)(wg1 + (kp));     \
    } while (0)
#define WAIT_STAGE()
#endif

    v8f acc[4][2];
#pragma unroll
    for (int mi = 0; mi < 4; ++mi)
#pragma unroll
        for (int ni = 0; ni < 2; ++ni)
            acc[mi][ni] = (v8f){0.f,0.f,0.f,0.f,0.f,0.f,0.f,0.f};

    // stage 0
    LOAD_STAGE(0, 0);
    WAIT_STAGE();
    __syncthreads();

    for (int kt = 0; kt < NK; ++kt) {
        const int  cur  = kt & 1;
        const bool more = (kt + 1) < NK;

        if (more) LOAD_STAGE(cur ^ 1, (kt + 1) * KT);   // overlaps with WMMAs
        if (kt + 2 < NK) {                               // distance-2 L2 prefetch
            const int kp2 = (kt + 2) * KT;
            __builtin_prefetch(xg0 + kp2, 0, 3);
            __builtin_prefetch(xg1 + kp2, 0, 3);
            __builtin_prefetch(wg0 + kp2, 0, 3);
            __builtin_prefetch(wg1 + kp2, 0, 3);
        }

        // 4 k-steps of 4, 8 WMMAs each = 32 v_wmma per wave per stage
#pragma unroll
        for (int kk = 0; kk < 4; ++kk) {
            const int kc = kk * 4 + 2 * half;   // lanes 0-15: K={0,1}; 16-31: K={2,3}
            v2f a[4], b[2];
#pragma unroll
            for (int mi = 0; mi < 4; ++mi)
                a[mi] = *(const v2f*)(&xs[cur][mBase + mi*16 + idx][kc]);
#pragma unroll
            for (int ni = 0; ni < 2; ++ni)
                b[ni] = *(const v2f*)(&ws[cur][nBase + ni*16 + idx][kc]);
#pragma unroll
            for (int mi = 0; mi < 4; ++mi)
#pragma unroll
                for (int ni = 0; ni < 2; ++ni)
                    acc[mi][ni] = __builtin_amdgcn_wmma_f32_16x16x4_f32(
                        false, a[mi], false, b[ni],
                        (short)0, acc[mi][ni], false, false);
        }

        if (more) WAIT_STAGE();   // async copies into the other buffer landed
        __syncthreads();
    }

    // Epilogue: D-fragment VGPR j -> M = j + 8*half, N = idx (ISA 16x16 f32 C/D).
    // y += bias[o] + dot16(u[t,:], lora_B[slot, o, :])  (lora_B = 2 MB, lives in L2)
#pragma unroll
    for (int ni = 0; ni < 2; ++ni) {
        const int   o  = oTile + nBase + ni * 16 + idx;
        const float bo = bias[o];
#pragma unroll
        for (int mi = 0; mi < 4; ++mi) {
#pragma unroll
            for (int j = 0; j < 8; ++j) {
                const int t = tTile + mBase + mi * 16 + j + 8 * half;
                float v = acc[mi][ni][j] + bo;
                const int slot = t2s[t];
                if (slot >= 0 && slot < NSLOT) {
                    const float4* up = (const float4*)(u + (size_t)t * RANK);
                    const float4* bp = (const float4*)(loraB +
                        ((size_t)slot * DOUT + o) * RANK);
#pragma unroll
                    for (int q = 0; q < 4; ++q) {
                        const float4 uv = up[q];
                        const float4 bv = bp[q];
                        v += uv.x*bv.x + uv.y*bv.y + uv.z*bv.z + uv.w*bv.w;
                    }
                }
                out[(size_t)t * DOUT + o] = v;
            }
        }
    }
}

// ---------------------------------------------------------------------------
extern "C" void kernel_launch(void* const* d_in, const int* in_sizes, int n_in,
                              void* d_out, int out_size, void* d_ws, size_t ws_size,
                              hipStream_t stream) {
    const float* x     = (const float*)d_in[0];
    const int*   t2s   = (const int*)  d_in[1];
    const float* wgt   = (const float*)d_in[2];
    const float* bias  = (const float*)d_in[3];
    const float* loraA = (const float*)d_in[4];
    const float* loraB = (const float*)d_in[5];
    const float* scal  = (const float*)d_in[6];
    float*       out   = (float*)d_out;
    float*       u     = (float*)d_ws;       // T*RANK fp32 = 512 KB scratch

    const int T = in_sizes[1];               // token count (8192)

    lora_u_kernel<<<T / 8, 256, 0, stream>>>(x, t2s, loraA, scal, u);

    dim3 grid(DOUT / BN, T / BM);
    lora_gemm_f32_wmma<<<grid, 256, 0, stream>>>(x, t2s, wgt, bias, loraB, u, out);
}